// Get_four_embedding_67765993997022
// MI455X (gfx1250) — compile-verified
//
#include <hip/hip_runtime.h>

#define MAX_SEP 512
#define T_ROWS 1025     // 2*MAX_SEP + 1
#define H_DIM 160
#define S_DIM 256
#define B_DIM 4

typedef __attribute__((ext_vector_type(2))) float v2f;
typedef __attribute__((ext_vector_type(4))) float v4f;
typedef __attribute__((ext_vector_type(8))) float v8f;

// ---------------------------------------------------------------------------
// Kernel 1: fold the Linear into the embedding tables.
//   Y_q[t, h] = sum_k pe_q[t, k] * W[h, q*H + k]
// One wave computes one 16x16 output tile via V_WMMA_F32_16X16X4_F32,
// looping K in steps of 4 (40 WMMAs per tile).
// ---------------------------------------------------------------------------
__global__ __launch_bounds__(256) void fold_tables_wmma(
    const float* __restrict__ pe_ss, const float* __restrict__ pe_se,
    const float* __restrict__ pe_es, const float* __restrict__ pe_ee,
    const float* __restrict__ W, float* __restrict__ Y)
{
    const int tid  = threadIdx.x;
    const int lane = tid & 31;
    const int wid  = blockIdx.x * (blockDim.x >> 5) + (tid >> 5);

    const int M_TILES = 65;   // ceil(1025 / 16)
    const int N_TILES = 10;   // 160 / 16
    if (wid >= 4 * M_TILES * N_TILES) return;   // wave-uniform guard

    const int q   = wid / (M_TILES * N_TILES);
    const int rem = wid % (M_TILES * N_TILES);
    const int mt  = rem / N_TILES;
    const int nt  = rem % N_TILES;

    const float* pe = (q == 0) ? pe_ss : (q == 1) ? pe_se
                    : (q == 2) ? pe_es : pe_ee;

    // A fragment (16x4 fp32): lane holds row m = lane&15;
    // VGPR0/VGPR1 = K pair; upper half-wave covers K+2.
    const int m    = lane & 15;
    const int koff = (lane >> 4) << 1;           // 0 or 2
    int arow = mt * 16 + m;
    if (arow > T_ROWS - 1) arow = T_ROWS - 1;    // clamp padded rows (load only)
    const float* aptr = pe + (size_t)arow * H_DIM + koff;

    // B fragment (4x16 fp32): B[k, n] = W[nt*16 + n, q*H + k]
    const int n = lane & 15;
    const float* bptr = W + (size_t)(nt * 16 + n) * (4 * H_DIM) + q * H_DIM + koff;

    v8f c = {};
    #pragma unroll 8
    for (int k0 = 0; k0 < H_DIM; k0 += 4) {
        v2f a = *(const v2f*)(aptr + k0);
        v2f b = *(const v2f*)(bptr + k0);
        // 8 args: (neg_a, A, neg_b, B, c_mod, C, reuse_a, reuse_b)
        c = __builtin_amdgcn_wmma_f32_16x16x4_f32(
                false, a, false, b, (short)0, c, false, false);
    }

    // C/D layout: lanes 0-15 -> rows 0..7 (VGPR idx), lanes 16-31 -> rows 8..15
    const int mbase = mt * 16 + ((lane >> 4) << 3);
    float* yp = Y + (size_t)q * (T_ROWS * H_DIM)
                  + (size_t)mbase * H_DIM + nt * 16 + n;

    if (mt < M_TILES - 1) {
        // Full tile: straight-line clause of 8 stores, EXEC untouched.
        #pragma unroll
        for (int v = 0; v < 8; ++v) yp[(size_t)v * H_DIM] = c[v];
    } else {
        // Tail tile (rows 1024..1039): only valid rows stored.
        #pragma unroll
        for (int v = 0; v < 8; ++v) {
            if (mbase + v < T_ROWS) yp[(size_t)v * H_DIM] = c[v];
        }
    }
}

// ---------------------------------------------------------------------------
// Kernel 2: out[b,i,j,h] = relu(Yss[dss,h] + Yse[dse,h] + Yes[des,h]
//                               + Yee[dee,h] + bias[h])
// One block per (b,i); pos rows cached in LDS; float4 per thread.
// Pure HBM-store-bound (~168 MB), gathers hit the L2-resident 2.6 MB tables.
// ---------------------------------------------------------------------------
__global__ __launch_bounds__(256) void gather_sum_relu(
    const int* __restrict__ pos_s, const int* __restrict__ pos_e,
    const float* __restrict__ Y, const float* __restrict__ bias,
    float* __restrict__ out)
{
    const int bi  = blockIdx.x;       // b*256 + i
    const int b   = bi >> 8;
    const int i   = bi & 255;
    const int tid = threadIdx.x;

    __shared__ int s_ps[S_DIM];
    __shared__ int s_pe[S_DIM];
    if (tid < S_DIM) {
        s_ps[tid] = pos_s[b * S_DIM + tid];
        s_pe[tid] = pos_e[b * S_DIM + tid];
    }
    __syncthreads();

    const int ps_i = s_ps[i];
    const int pe_i = s_pe[i];

    const float* Yss = Y;
    const float* Yse = Y + 1 * (size_t)T_ROWS * H_DIM;
    const float* Yes = Y + 2 * (size_t)T_ROWS * H_DIM;
    const float* Yee = Y + 3 * (size_t)T_ROWS * H_DIM;

    float* outrow = out + (size_t)bi * (S_DIM * H_DIM);

    const int NQ4 = S_DIM * (H_DIM / 4);          // 10240 float4 per block
    for (int q4 = tid; q4 < NQ4; q4 += 256) {
        const int j = q4 / (H_DIM / 4);
        const int h = (q4 % (H_DIM / 4)) * 4;
        const int ps_j = s_ps[j];
        const int pe_j = s_pe[j];
        const int dss = ps_i - ps_j + MAX_SEP;
        const int dse = ps_i - pe_j + MAX_SEP;
        const int des = pe_i - ps_j + MAX_SEP;
        const int dee = pe_i - pe_j + MAX_SEP;

        v4f acc = *(const v4f*)(Yss + (size_t)dss * H_DIM + h);
        acc    += *(const v4f*)(Yse + (size_t)dse * H_DIM + h);
        acc    += *(const v4f*)(Yes + (size_t)des * H_DIM + h);
        acc    += *(const v4f*)(Yee + (size_t)dee * H_DIM + h);
        acc    += *(const v4f*)(bias + h);
        #pragma unroll
        for (int v = 0; v < 4; ++v) acc[v] = fmaxf(acc[v], 0.0f);

        *(v4f*)(outrow + (size_t)j * H_DIM + h) = acc;
    }
}

// ---------------------------------------------------------------------------
extern "C" void kernel_launch(void* const* d_in, const int* in_sizes, int n_in,
                              void* d_out, int out_size, void* d_ws, size_t ws_size,
                              hipStream_t stream)
{
    const int*   pos_s = (const int*)  d_in[0];
    const int*   pos_e = (const int*)  d_in[1];
    const float* pe_ss = (const float*)d_in[2];
    const float* pe_se = (const float*)d_in[3];
    const float* pe_es = (const float*)d_in[4];
    const float* pe_ee = (const float*)d_in[5];
    const float* W     = (const float*)d_in[6];
    const float* bias  = (const float*)d_in[7];

    float* Y   = (float*)d_ws;   // 4 * 1025 * 160 floats = 2.6 MB
    float* out = (float*)d_out;

    const int waves   = 4 * 65 * 10;            // tiles: q * M_tiles * N_tiles
    const int blocks1 = (waves + 7) / 8;        // 8 waves / 256-thread block

    fold_tables_wmma<<<blocks1, 256, 0, stream>>>(pe_ss, pe_se, pe_es, pe_ee, W, Y);
    gather_sum_relu<<<B_DIM * S_DIM, 256, 0, stream>>>(pos_s, pos_e, Y, bias, out);
}